// VQVAE_45861660786778
// MI455X (gfx1250) — compile-verified
//
#include <hip/hip_runtime.h>
#include <math.h>

// VQ-VAE codebook quantization for MI455X (gfx1250, wave32).
//
// Memory-bound: 128 MB traffic -> ~5.5us floor @ 23.3 TB/s. The distance
// matmul  score = ||c||^2 - 2 z.c  is computed on the matrix path with
// chained V_WMMA_F32_16X16X4_F32, with ||c||^2 pre-baked into the C
// accumulator so no VALU post-processing of the WMMA result is needed.
//
// Per 16-vector tile (one wave):
//   A (loop-invariant) = -2 * codebook rows, M = code index   (2x v2f per half)
//   B = z-tile transposed, N = vector index within tile       (2x global_load_b64)
//   D0 = wmma(a0_lo, b0, cnorm_lo); D0 = wmma(a1_lo, b1, D0)  (codes 0..15)
//   D1 = wmma(a0_hi, b0, cnorm_hi); D1 = wmma(a1_hi, b1, D1)  (codes 16..31)
//   argmin: per-lane over 16 scores, then shfl_xor(16) combine (lane n and
//           n+16 hold complementary halves of the 32 candidates for vector n).
//   z_q row written as two b128 halves gathered from LDS codebook.

typedef __attribute__((ext_vector_type(2))) float v2f;
typedef __attribute__((ext_vector_type(8))) float v8f;

#define TPB        256
#define WAVES_PB   (TPB / 32)
#define NBLOCKS    512
#define KCODES     32
#define DDIM       8
#define COMMIT     0.25f

__global__ __launch_bounds__(TPB) void vq_main_kernel(
    const float* __restrict__ z,
    const float* __restrict__ cb,
    float* __restrict__ out,
    float* __restrict__ partials,        // [NBLOCKS] deterministic loss partials
    unsigned int* __restrict__ counts,   // [32] global histogram
    int N)
{
    __shared__ __align__(16) float s_cb[KCODES * DDIM];   // 1 KB codebook
    __shared__ float        s_cn[KCODES];                  // ||c||^2
    __shared__ unsigned int s_hist[KCODES];
    __shared__ float        s_wsum[WAVES_PB];

    const int tid = threadIdx.x;

    // Stage codebook + squared norms + clear histogram.
    if (tid < KCODES * DDIM) s_cb[tid] = cb[tid];
    if (tid < KCODES) s_hist[tid] = 0u;
    __syncthreads();
    if (tid < KCODES) {
        float s = 0.f;
        #pragma unroll
        for (int k = 0; k < DDIM; ++k) { float v = s_cb[tid * DDIM + k]; s += v * v; }
        s_cn[tid] = s;
    }
    __syncthreads();

    const int lane  = tid & 31;
    const int wv    = tid >> 5;
    const int half  = lane >> 4;      // 0: lanes 0-15, 1: lanes 16-31
    const int l16   = lane & 15;
    const int base0 = half * 8;       // M offset inside a D fragment
    const int k0    = half * 2;       // A/B K-split: lanes<16 -> K 0,1 ; lanes>=16 -> K 2,3

    // Loop-invariant A fragments: A = -2 * codebook  (16x4 f32 layout:
    // lane = M (mod 16), VGPR0/1 = K k0,k0+1).
    v2f a0_lo, a1_lo, a0_hi, a1_hi;
    {
        const float* rlo = s_cb + l16 * DDIM;
        const float* rhi = s_cb + (16 + l16) * DDIM;
        a0_lo.x = -2.f * rlo[k0];     a0_lo.y = -2.f * rlo[k0 + 1];
        a1_lo.x = -2.f * rlo[k0 + 4]; a1_lo.y = -2.f * rlo[k0 + 5];
        a0_hi.x = -2.f * rhi[k0];     a0_hi.y = -2.f * rhi[k0 + 1];
        a1_hi.x = -2.f * rhi[k0 + 4]; a1_hi.y = -2.f * rhi[k0 + 5];
    }

    // Loop-invariant C accumulators carrying ||c||^2 per (lane, VGPR) element.
    v8f c_lo, c_hi;
    #pragma unroll
    for (int r = 0; r < 8; ++r) {
        c_lo[r] = s_cn[base0 + r];
        c_hi[r] = s_cn[16 + base0 + r];
    }

    float acc = 0.f;
    const int numTiles = N >> 4;
    const int stride   = gridDim.x * WAVES_PB;

    for (int t = blockIdx.x * WAVES_PB + wv; t < numTiles; t += stride) {
        const int n = (t << 4) + l16;                 // z-vector owned by this lane pair
        const float* zr = z + (size_t)n * DDIM;

        // B fragment: z-tile transposed (lane = N, VGPRs = K k0,k0+1 then k0+4,k0+5)
        v2f b0 = *(const v2f*)(zr + k0);
        v2f b1 = *(const v2f*)(zr + k0 + 4);

        // score = ||c||^2 - 2 z.c   via chained f32 WMMAs (K = 8 in two k=4 steps)
        v8f d0 = __builtin_amdgcn_wmma_f32_16x16x4_f32(false, a0_lo, false, b0,
                                                       (short)0, c_lo, false, false);
        d0     = __builtin_amdgcn_wmma_f32_16x16x4_f32(false, a1_lo, false, b1,
                                                       (short)0, d0, false, false);
        v8f d1 = __builtin_amdgcn_wmma_f32_16x16x4_f32(false, a0_hi, false, b0,
                                                       (short)0, c_hi, false, false);
        d1     = __builtin_amdgcn_wmma_f32_16x16x4_f32(false, a1_hi, false, b1,
                                                       (short)0, d1, false, false);

        // Per-lane argmin over its 16 candidates (indices ascending -> first-
        // occurrence semantics preserved with strict <).
        float best = d0[0];
        int   bidx = base0;
        #pragma unroll
        for (int r = 1; r < 8; ++r)
            if (d0[r] < best) { best = d0[r]; bidx = base0 + r; }
        #pragma unroll
        for (int r = 0; r < 8; ++r)
            if (d1[r] < best) { best = d1[r]; bidx = 16 + base0 + r; }

        // Combine with xor-16 partner (holds the other 16 candidates for the
        // same vector). Tie-break on smaller index to match jnp.argmin.
        float obest = __shfl_xor(best, 16, 32);
        int   oidx  = __shfl_xor(bidx, 16, 32);
        if (obest < best || (obest == best && oidx < bidx)) { best = obest; bidx = oidx; }

        // z_q_st row == codebook[bidx]; the lane pair writes its two b128 halves.
        const float4* cbv = (const float4*)s_cb;
        float4 q = cbv[bidx * 2 + half];
        *(float4*)(out + (size_t)n * DDIM + half * 4) = q;

        // Loss: this lane owns elements {k0, k0+1, k0+4, k0+5} of vector n.
        const float* cq = s_cb + bidx * DDIM;
        float e0 = b0.x - cq[k0];
        float e1 = b0.y - cq[k0 + 1];
        float e2 = b1.x - cq[k0 + 4];
        float e3 = b1.y - cq[k0 + 5];
        acc += e0 * e0 + e1 * e1 + e2 * e2 + e3 * e3;

        // One histogram vote per vector (lanes 0-15 only).
        if (!half) atomicAdd(&s_hist[bidx], 1u);
    }

    // Deterministic loss reduction: wave shuffles -> LDS -> per-block partial.
    #pragma unroll
    for (int off = 16; off > 0; off >>= 1) acc += __shfl_down(acc, off, 32);
    if (lane == 0) s_wsum[wv] = acc;
    __syncthreads();
    if (tid == 0) {
        float s = 0.f;
        #pragma unroll
        for (int i = 0; i < WAVES_PB; ++i) s += s_wsum[i];
        partials[blockIdx.x] = s;
    }
    if (tid < KCODES) atomicAdd(&counts[tid], s_hist[tid]);
}

__global__ void vq_finalize_kernel(const float* __restrict__ partials,
                                   const unsigned int* __restrict__ counts,
                                   float* __restrict__ out, int N, int nblocks)
{
    if (threadIdx.x == 0 && blockIdx.x == 0) {
        float s = 0.f;
        for (int i = 0; i < nblocks; ++i) s += partials[i];   // fixed order
        float vq_loss = (1.f + COMMIT) * s / ((float)N * (float)DDIM);

        float H = 0.f;
        for (int k = 0; k < KCODES; ++k) {
            float p = (float)counts[k] / (float)N;
            H += p * logf(p + 1e-10f);
        }
        out[(size_t)N * DDIM]     = vq_loss;
        out[(size_t)N * DDIM + 1] = expf(-H);
    }
}

extern "C" void kernel_launch(void* const* d_in, const int* in_sizes, int n_in,
                              void* d_out, int out_size, void* d_ws, size_t ws_size,
                              hipStream_t stream) {
    const float* z  = (const float*)d_in[0];
    const float* cb = (const float*)d_in[1];
    float* out = (float*)d_out;
    const int N = in_sizes[0] / DDIM;

    unsigned int* counts = (unsigned int*)d_ws;                 // 32 * 4B
    float* partials      = (float*)((char*)d_ws + 128);         // NBLOCKS * 4B

    hipMemsetAsync(d_ws, 0, 128, stream);                       // zero histogram
    vq_main_kernel<<<NBLOCKS, TPB, 0, stream>>>(z, cb, out, partials, counts, N);
    vq_finalize_kernel<<<1, 32, 0, stream>>>(partials, counts, out, N, NBLOCKS);
}